// CrossMultiHeadAttention_34565896798834
// MI455X (gfx1250) — compile-verified
//
#include <hip/hip_runtime.h>
#include <math.h>

// ---------------------------------------------------------------------------
// CrossMultiHeadAttention (criss-cross attention) for MI455X / gfx1250.
// Memory-bound (268MB energy output dominates); all GEMMs use
// V_WMMA_F32_16X16X4_F32 to keep full fp32 precision at zero perf cost.
// Requires ws_size >= ~75 MB.
// ---------------------------------------------------------------------------

typedef __attribute__((ext_vector_type(2))) float v2f;
typedef __attribute__((ext_vector_type(8))) float v8f;

#define B_      4
#define C_      128
#define N_      128
#define NN_     (N_ * N_)
#define HEADS_  4
#define GROUPS_ 4
#define DQ_     64
#define DHQ_    16      // q/k channels per head
#define DHV_    32      // v channels per head
#define EK_     256     // energy last-dim (2*N)
#define EPS_    1e-5f

// workspace offsets (in floats)
#define WS_MU    0
#define WS_VAR   16
#define WS_WEFF  32                                   // B*256*C
#define WS_BIAS  (WS_WEFF + B_ * 256 * C_)            // B*256
#define WS_QKV   (WS_BIAS + B_ * 256)                 // B*256*NN
#define WS_HSTAT (WS_QKV + (size_t)B_ * 256 * NN_)    // B*H*NN*4 (maxH,sumH,maxW,sumW)
#define WS_RSTAT (WS_HSTAT + (size_t)B_ * HEADS_ * NN_ * 4) // B*H*NN*2 (max, 1/sum)

// ---------------- WMMA helpers (wave32, 16x16x4 f32) -----------------------

__device__ __forceinline__ v8f wmma4(v2f a, v2f b, v8f c) {
    // (neg_a, A, neg_b, B, c_mod, C, reuse_a, reuse_b)
    return __builtin_amdgcn_wmma_f32_16x16x4_f32(false, a, false, b, (short)0, c,
                                                 false, false);
}

// A(16x4) fragment from row-major LDS matrix S[m][k], leading dim ld.
__device__ __forceinline__ v2f loadA(const float* s, int ld, int m0, int k0, int lane) {
    int m = m0 + (lane & 15);
    int k = k0 + ((lane >> 4) << 1);
    v2f a;
    a.x = s[m * ld + k];
    a.y = s[m * ld + k + 1];
    return a;
}

// A(16x4) fragment where the logical A[m][k] is stored transposed: S[k][m].
__device__ __forceinline__ v2f loadAT(const float* s, int ld, int m0, int k0, int lane) {
    int m = m0 + (lane & 15);
    int k = k0 + ((lane >> 4) << 1);
    v2f a;
    a.x = s[k * ld + m];
    a.y = s[(k + 1) * ld + m];
    return a;
}

// B(4x16) fragment from row-major LDS matrix S[k][n], leading dim ld.
__device__ __forceinline__ v2f loadB(const float* s, int ld, int k0, int n0, int lane) {
    int n = n0 + (lane & 15);
    int k = k0 + ((lane >> 4) << 1);
    v2f b;
    b.x = s[k * ld + n];
    b.y = s[(k + 1) * ld + n];
    return b;
}

// B(4x16) fragment where the logical B[k][n] is stored transposed: S[n][k].
__device__ __forceinline__ v2f loadBT(const float* s, int ld, int k0, int n0, int lane) {
    int n = n0 + (lane & 15);
    int k = k0 + ((lane >> 4) << 1);
    v2f b;
    b.x = s[n * ld + k];
    b.y = s[n * ld + k + 1];
    return b;
}

__device__ __forceinline__ v8f vzero8() {
    v8f z = {0.f, 0.f, 0.f, 0.f, 0.f, 0.f, 0.f, 0.f};
    return z;
}

// ---------------- K1: group-norm statistics --------------------------------
// One block per (b, g): reduce 32*NN contiguous floats.
__global__ void k_gn_stats(const float* __restrict__ x, float* __restrict__ ws) {
    int bg = blockIdx.x;  // b*4+g
    const float* p = x + (size_t)bg * (C_ / GROUPS_) * NN_;
    const int n = (C_ / GROUPS_) * NN_;  // 524288
    float s = 0.f, s2 = 0.f;
    for (int i = threadIdx.x; i < n / 4; i += blockDim.x) {
        float4 v = ((const float4*)p)[i];
        s += v.x + v.y + v.z + v.w;
        s2 += v.x * v.x + v.y * v.y + v.z * v.z + v.w * v.w;
    }
    __shared__ float rs[256], rq[256];
    rs[threadIdx.x] = s;
    rq[threadIdx.x] = s2;
    __syncthreads();
    for (int st = 128; st > 0; st >>= 1) {
        if ((int)threadIdx.x < st) {
            rs[threadIdx.x] += rs[threadIdx.x + st];
            rq[threadIdx.x] += rq[threadIdx.x + st];
        }
        __syncthreads();
    }
    if (threadIdx.x == 0) {
        float mu = rs[0] / (float)n;
        float var = rq[0] / (float)n - mu * mu;
        ws[WS_MU + bg] = mu;
        ws[WS_VAR + bg] = var;
    }
}

// ---------------- K2: fold norm into projection weights --------------------
// One block per (b, o), 128 threads (c).
__global__ void k_weff(const float* __restrict__ wqkv, const float* __restrict__ gamma,
                       const float* __restrict__ beta, float* __restrict__ ws) {
    int bo = blockIdx.x;
    int b = bo >> 8, o = bo & 255;
    int c = threadIdx.x;
    int g = c >> 5;
    float mu = ws[WS_MU + b * GROUPS_ + g];
    float var = ws[WS_VAR + b * GROUPS_ + g];
    float rstd = rsqrtf(var + EPS_);
    float sc = rstd * gamma[c];
    float tc = beta[c] - mu * sc;
    float w = wqkv[o * C_ + c];
    ws[WS_WEFF + ((size_t)b * 256 + o) * C_ + c] = w * sc;
    __shared__ float red[128];
    red[c] = w * tc;
    __syncthreads();
    for (int st = 64; st > 0; st >>= 1) {
        if (c < st) red[c] += red[c + st];
        __syncthreads();
    }
    if (c == 0) ws[WS_BIAS + b * 256 + o] = red[0];
}

// ---------------- K3: QKV GEMM (256x128)x(128x16384) per batch -------------
// grid (ntile=128, mhalf=2, b=4), 256 threads = 8 waves.
__global__ void k_qkv(const float* __restrict__ x, float* __restrict__ ws) {
    int nt = blockIdx.x, mh = blockIdx.y, b = blockIdx.z;
    int p0 = nt * 128;
    int o0 = mh * 128;
    const float* A = ws + WS_WEFF + ((size_t)b * 256 + o0) * C_;  // 128(o) x 128(c)
    const float* X = x + (size_t)b * C_ * NN_;                    // 128(c) x NN(p)
    float* Q = ws + WS_QKV + (size_t)b * 256 * NN_;
    const float* bias = ws + WS_BIAS + b * 256;

    __shared__ float As[128][33];  // o x c-slab
    __shared__ float Bs[32][129];  // c-slab x p
    int lane = threadIdx.x & 31, wave = threadIdx.x >> 5;

    v8f acc[8];
#pragma unroll
    for (int i = 0; i < 8; ++i) acc[i] = vzero8();

    for (int kt = 0; kt < 4; ++kt) {
        int c0 = kt * 32;
        for (int i = threadIdx.x; i < 128 * 32; i += 256) {
            int r = i >> 5, cc = i & 31;
            As[r][cc] = A[r * C_ + c0 + cc];
        }
        for (int i = threadIdx.x; i < 32 * 128; i += 256) {
            int r = i >> 7, cc = i & 127;
            Bs[r][cc] = X[(size_t)(c0 + r) * NN_ + p0 + cc];
        }
        __syncthreads();
        int m0 = wave * 16;
#pragma unroll
        for (int kk = 0; kk < 8; ++kk) {
            v2f a = loadA(&As[0][0], 33, m0, kk * 4, lane);
#pragma unroll
            for (int t = 0; t < 8; ++t) {
                v2f bb = loadB(&Bs[0][0], 129, kk * 4, t * 16, lane);
                acc[t] = wmma4(a, bb, acc[t]);
            }
        }
        __syncthreads();
    }
    int m0 = wave * 16;
    int col = lane & 15, hi8 = (lane >> 4) << 3;
#pragma unroll
    for (int t = 0; t < 8; ++t) {
#pragma unroll
        for (int r = 0; r < 8; ++r) {
            int row = o0 + m0 + hi8 + r;
            Q[(size_t)row * NN_ + p0 + t * 16 + col] = acc[t][r] + bias[row];
        }
    }
}

// ---------------- K4a: eH energy (attention along H, fixed w) --------------
// grid (w=128, n=4, b=4); E[h][k] = sum_c q[c,h,w]*k[c,kidx,w]; diag=-inf; mask.
__global__ void k_eH(const int* __restrict__ mask, float* __restrict__ ws,
                     float* __restrict__ energy) {
    int w = blockIdx.x, n = blockIdx.y, b = blockIdx.z;
    const float* qkv = ws + WS_QKV + (size_t)b * 256 * NN_;
    __shared__ float Qs[16][129];
    __shared__ float Ks[16][129];
    __shared__ float E[64][129];
    for (int i = threadIdx.x; i < 16 * 128; i += 256) {
        int c = i >> 7, h = i & 127;
        Qs[c][h] = qkv[(size_t)(DHQ_ * n + c) * NN_ + h * N_ + w];
        Ks[c][h] = qkv[(size_t)(DQ_ + DHQ_ * n + c) * NN_ + h * N_ + w];
    }
    __syncthreads();
    int lane = threadIdx.x & 31, wave = threadIdx.x >> 5;
    int col = lane & 15, hi8 = (lane >> 4) << 3;

    for (int half = 0; half < 2; ++half) {
        int h0 = half * 64;
        // 4 m-tiles x 8 n-tiles; wave owns column tile `wave`, loops m-tiles.
#pragma unroll
        for (int mt = 0; mt < 4; ++mt) {
            v8f acc = vzero8();
#pragma unroll
            for (int kk = 0; kk < 4; ++kk) {
                v2f a = loadAT(&Qs[0][0], 129, h0 + mt * 16, kk * 4, lane);
                v2f bb = loadB(&Ks[0][0], 129, kk * 4, wave * 16, lane);
                acc = wmma4(a, bb, acc);
            }
#pragma unroll
            for (int r = 0; r < 8; ++r) E[mt * 16 + hi8 + r][wave * 16 + col] = acc[r];
        }
        __syncthreads();
        if (threadIdx.x < 64) {
            int hl = threadIdx.x;
            int h = h0 + hl;
            E[hl][h] = -INFINITY;  // diagonal h == k
            int mv = mask[(b * N_ + h) * N_ + w];
            float m = -INFINITY, ssum = 0.f;
            if (mv == 0) {
                for (int k = 0; k < 128; ++k) E[hl][k] = -1e20f;
                m = -1e20f;
                ssum = 128.f;
            } else {
                for (int k = 0; k < 128; ++k) m = fmaxf(m, E[hl][k]);
                for (int k = 0; k < 128; ++k) ssum += __expf(E[hl][k] - m);
            }
            float* hs = ws + WS_HSTAT + ((size_t)(((b * HEADS_ + n) * N_ + h) * N_ + w)) * 4;
            hs[0] = m;
            hs[1] = ssum;
        }
        __syncthreads();
        for (int i = threadIdx.x; i < 64 * 128; i += 256) {
            int hl = i >> 7, k = i & 127;
            size_t row = (size_t)(((b * HEADS_ + n) * N_ + h0 + hl) * N_ + w);
            energy[row * EK_ + k] = E[hl][k];
        }
        __syncthreads();
    }
}

// ---------------- K4b: eW energy (attention along W, fixed h) --------------
__global__ void k_eW(const int* __restrict__ mask, float* __restrict__ ws,
                     float* __restrict__ energy) {
    int h = blockIdx.x, n = blockIdx.y, b = blockIdx.z;
    const float* qkv = ws + WS_QKV + (size_t)b * 256 * NN_;
    __shared__ float Qs[16][129];
    __shared__ float Ks[16][129];
    __shared__ float E[64][129];
    for (int i = threadIdx.x; i < 16 * 128; i += 256) {
        int c = i >> 7, w = i & 127;
        Qs[c][w] = qkv[(size_t)(DHQ_ * n + c) * NN_ + h * N_ + w];
        Ks[c][w] = qkv[(size_t)(DQ_ + DHQ_ * n + c) * NN_ + h * N_ + w];
    }
    __syncthreads();
    int lane = threadIdx.x & 31, wave = threadIdx.x >> 5;
    int col = lane & 15, hi8 = (lane >> 4) << 3;

    for (int half = 0; half < 2; ++half) {
        int w0 = half * 64;
#pragma unroll
        for (int mt = 0; mt < 4; ++mt) {
            v8f acc = vzero8();
#pragma unroll
            for (int kk = 0; kk < 4; ++kk) {
                v2f a = loadAT(&Qs[0][0], 129, w0 + mt * 16, kk * 4, lane);
                v2f bb = loadB(&Ks[0][0], 129, kk * 4, wave * 16, lane);
                acc = wmma4(a, bb, acc);
            }
#pragma unroll
            for (int r = 0; r < 8; ++r) E[mt * 16 + hi8 + r][wave * 16 + col] = acc[r];
        }
        __syncthreads();
        if (threadIdx.x < 64) {
            int wl = threadIdx.x;
            int w = w0 + wl;
            int mv = mask[(b * N_ + h) * N_ + w];
            float m = -INFINITY, ssum = 0.f;
            if (mv == 0) {
                for (int k = 0; k < 128; ++k) E[wl][k] = -1e20f;
                m = -1e20f;
                ssum = 128.f;
            } else {
                for (int k = 0; k < 128; ++k) m = fmaxf(m, E[wl][k]);
                for (int k = 0; k < 128; ++k) ssum += __expf(E[wl][k] - m);
            }
            float* hs = ws + WS_HSTAT + ((size_t)(((b * HEADS_ + n) * N_ + h) * N_ + w)) * 4;
            hs[2] = m;
            hs[3] = ssum;
        }
        __syncthreads();
        for (int i = threadIdx.x; i < 64 * 128; i += 256) {
            int wl = i >> 7, k = i & 127;
            size_t row = (size_t)(((b * HEADS_ + n) * N_ + h) * N_ + w0 + wl);
            energy[row * EK_ + 128 + k] = E[wl][k];
        }
        __syncthreads();
    }
}

// ---------------- K5: merge half softmax stats (log-sum-exp) ---------------
__global__ void k_rowstats(float* __restrict__ ws) {
    int i = blockIdx.x * blockDim.x + threadIdx.x;  // 0 .. B*H*NN-1
    const float* hs = ws + WS_HSTAT + (size_t)i * 4;
    float mH = hs[0], sH = hs[1], mW = hs[2], sW = hs[3];
    float m = fmaxf(mH, mW);
    float s = sH * __expf(mH - m) + sW * __expf(mW - m);
    float* rs = ws + WS_RSTAT + (size_t)i * 2;
    rs[0] = m;
    rs[1] = 1.0f / s;
}

// ---------------- K6: oH apply + x residual (writes out) -------------------
// grid (w=128, n=4, b=4). out[c,h,w] = x + sum_k v[c,k,w]*aH[h,w,k]
__global__ void k_oH(const float* __restrict__ x, float* __restrict__ ws,
                     const float* __restrict__ energy, float* __restrict__ out) {
    int w = blockIdx.x, n = blockIdx.y, b = blockIdx.z;
    const float* qkv = ws + WS_QKV + (size_t)b * 256 * NN_;
    const float* rs = ws + WS_RSTAT;
    __shared__ float V[32][129];
    __shared__ float Ah[64][129];
    for (int i = threadIdx.x; i < 32 * 128; i += 256) {
        int c = i >> 7, k = i & 127;
        V[c][k] = qkv[(size_t)(C_ + DHV_ * n + c) * NN_ + k * N_ + w];
    }
    int lane = threadIdx.x & 31, wave = threadIdx.x >> 5;
    int col = lane & 15, hi8 = (lane >> 4) << 3;
    int mt = wave >> 2, ntl = wave & 3;

    for (int half = 0; half < 2; ++half) {
        int h0 = half * 64;
        __syncthreads();  // V ready / Ah free
        for (int i = threadIdx.x; i < 64 * 128; i += 256) {
            int hl = i >> 7, k = i & 127;
            size_t row = (size_t)(((b * HEADS_ + n) * N_ + h0 + hl) * N_ + w);
            float m = rs[row * 2], is = rs[row * 2 + 1];
            Ah[hl][k] = __expf(energy[row * EK_ + k] - m) * is;
        }
        __syncthreads();
        v8f acc = vzero8();
#pragma unroll
        for (int kk = 0; kk < 32; ++kk) {
            v2f a = loadA(&V[0][0], 129, mt * 16, kk * 4, lane);
            v2f bb = loadBT(&Ah[0][0], 129, kk * 4, ntl * 16, lane);
            acc = wmma4(a, bb, acc);
        }
#pragma unroll
        for (int r = 0; r < 8; ++r) {
            int c = mt * 16 + hi8 + r;
            int h = h0 + ntl * 16 + col;
            size_t oi = ((size_t)((b * C_ + DHV_ * n + c) * N_ + h)) * N_ + w;
            out[oi] = x[oi] + acc[r];
        }
    }
}

// ---------------- K7: oW apply (accumulates into out) ----------------------
// grid (h=128, n=4, b=4). out[c,h,w] += sum_k v[c,h,k]*aW[h,w,k]
__global__ void k_oW(float* __restrict__ ws, const float* __restrict__ energy,
                     float* __restrict__ out) {
    int h = blockIdx.x, n = blockIdx.y, b = blockIdx.z;
    const float* qkv = ws + WS_QKV + (size_t)b * 256 * NN_;
    const float* rs = ws + WS_RSTAT;
    __shared__ float V[32][129];
    __shared__ float Aw[64][129];
    for (int i = threadIdx.x; i < 32 * 128; i += 256) {
        int c = i >> 7, k = i & 127;
        V[c][k] = qkv[(size_t)(C_ + DHV_ * n + c) * NN_ + h * N_ + k];
    }
    int lane = threadIdx.x & 31, wave = threadIdx.x >> 5;
    int col = lane & 15, hi8 = (lane >> 4) << 3;
    int mt = wave >> 2, ntl = wave & 3;

    for (int half = 0; half < 2; ++half) {
        int w0 = half * 64;
        __syncthreads();
        for (int i = threadIdx.x; i < 64 * 128; i += 256) {
            int wl = i >> 7, k = i & 127;
            size_t row = (size_t)(((b * HEADS_ + n) * N_ + h) * N_ + w0 + wl);
            float m = rs[row * 2], is = rs[row * 2 + 1];
            Aw[wl][k] = __expf(energy[row * EK_ + 128 + k] - m) * is;
        }
        __syncthreads();
        v8f acc = vzero8();
#pragma unroll
        for (int kk = 0; kk < 32; ++kk) {
            v2f a = loadA(&V[0][0], 129, mt * 16, kk * 4, lane);
            v2f bb = loadBT(&Aw[0][0], 129, kk * 4, ntl * 16, lane);
            acc = wmma4(a, bb, acc);
        }
#pragma unroll
        for (int r = 0; r < 8; ++r) {
            int c = mt * 16 + hi8 + r;
            int w = w0 + ntl * 16 + col;
            size_t oi = ((size_t)((b * C_ + DHV_ * n + c) * N_ + h)) * N_ + w;
            out[oi] += acc[r];  // unique ownership per element in this kernel
        }
    }
}

// ---------------------------------------------------------------------------

extern "C" void kernel_launch(void* const* d_in, const int* in_sizes, int n_in,
                              void* d_out, int out_size, void* d_ws, size_t ws_size,
                              hipStream_t stream) {
    (void)in_sizes; (void)n_in; (void)out_size; (void)ws_size;
    const float* x = (const float*)d_in[0];
    const int* mask = (const int*)d_in[1];
    const float* gamma = (const float*)d_in[2];
    const float* beta = (const float*)d_in[3];
    const float* wqkv = (const float*)d_in[4];
    float* out = (float*)d_out;
    float* energy = out + (size_t)B_ * C_ * NN_;
    float* ws = (float*)d_ws;

    k_gn_stats<<<dim3(B_ * GROUPS_), dim3(256), 0, stream>>>(x, ws);
    k_weff<<<dim3(B_ * 256), dim3(128), 0, stream>>>(wqkv, gamma, beta, ws);
    k_qkv<<<dim3(128, 2, B_), dim3(256), 0, stream>>>(x, ws);
    k_eH<<<dim3(N_, HEADS_, B_), dim3(256), 0, stream>>>(mask, ws, energy);
    k_eW<<<dim3(N_, HEADS_, B_), dim3(256), 0, stream>>>(mask, ws, energy);
    k_rowstats<<<dim3((B_ * HEADS_ * NN_) / 256), dim3(256), 0, stream>>>(ws);
    k_oH<<<dim3(N_, HEADS_, B_), dim3(256), 0, stream>>>(x, ws, energy, out);
    k_oW<<<dim3(N_, HEADS_, B_), dim3(256), 0, stream>>>(ws, energy, out);
}